// GPUTimeMask_38010460570421
// MI455X (gfx1250) — compile-verified
//
#include <hip/hip_runtime.h>
#include <hip/hip_bf16.h>

// Streaming time-mask: out[b,c,t] = x[b,c,t] * !(t in [s0,e0) or [s1,e1)) per batch b.
// Memory-bound: ~491 MB traffic, ~21 us floor at 23.3 TB/s. Optimized for gfx1250:
// wave32 blocks, 128-bit global ops, non-temporal hints (working set > 192MB L2),
// scalar (SGPR) mask parameters + uniform branch to skip per-lane compares on the
// >99% of tiles that touch no mask interval.

#define MAX_MASK_WIDTH 150
#define T_DIM      60000
#define C_DIM      16
#define B_DIM      64
#define T4_PER_ROW (T_DIM / 4)          // 15000 float4 per (b,c) row
#define BLOCK_THREADS 256               // 8 wave32 waves
#define T4_PER_BLOCK  BLOCK_THREADS     // one float4 per thread
#define BLOCKS_PER_ROW ((T4_PER_ROW + T4_PER_BLOCK - 1) / T4_PER_BLOCK)  // 59

typedef float v4f __attribute__((ext_vector_type(4)));

__global__ __launch_bounds__(BLOCK_THREADS)
void GPUTimeMask_38010460570421_kernel(const float* __restrict__ x,
                                       const int*   __restrict__ starts,
                                       const int*   __restrict__ widths,
                                       float*       __restrict__ out) {
    const int b  = blockIdx.z;                       // uniform -> scalar regs
    const int c  = blockIdx.y;
    const int t4 = blockIdx.x * T4_PER_BLOCK + threadIdx.x;
    if (t4 >= T4_PER_ROW) return;

    // Mask parameters: uniform per block -> s_load_b32 + SALU.
    int s0 = starts[b];
    int s1 = starts[B_DIM + b];
    int w0 = widths[b];
    int w1 = widths[B_DIM + b];
    w0 = min(max(w0, 1), MAX_MASK_WIDTH);
    w1 = min(max(w1, 1), MAX_MASK_WIDTH);
    const int e0 = s0 + w0;
    const int e1 = s1 + w1;

    const long row = ((long)b * C_DIM + c) * (long)T4_PER_ROW;
    const v4f* __restrict__ src = (const v4f*)x + row + t4;
    v4f*       __restrict__ dst = (v4f*)out + row + t4;

    // Single-use streaming data: non-temporal 128-bit load/store.
    v4f v = __builtin_nontemporal_load(src);

    // Uniform tile-overlap test (scalar branch). Tile covers 1024 time steps.
    const int tile_lo = blockIdx.x * (T4_PER_BLOCK * 4);
    const int tile_hi = tile_lo + (T4_PER_BLOCK * 4);
    const bool tile_overlaps =
        ((tile_lo < e0) & (tile_hi > s0)) | ((tile_lo < e1) & (tile_hi > s1));

    if (tile_overlaps) {
        const int t = t4 * 4;
#pragma unroll
        for (int k = 0; k < 4; ++k) {
            const int tk = t + k;
            const bool masked = ((tk >= s0) & (tk < e0)) | ((tk >= s1) & (tk < e1));
            v[k] = masked ? 0.0f : v[k];
        }
    }

    __builtin_nontemporal_store(v, dst);
}

extern "C" void kernel_launch(void* const* d_in, const int* in_sizes, int n_in,
                              void* d_out, int out_size, void* d_ws, size_t ws_size,
                              hipStream_t stream) {
    const float* x      = (const float*)d_in[0];   // (64, 16, 60000) fp32
    const int*   starts = (const int*)d_in[1];     // (2, 64) int32
    const int*   widths = (const int*)d_in[2];     // (2, 64) int32
    float*       out    = (float*)d_out;           // (64, 16, 60000) fp32

    dim3 grid(BLOCKS_PER_ROW, C_DIM, B_DIM);       // 59 x 16 x 64 blocks
    dim3 block(BLOCK_THREADS);
    GPUTimeMask_38010460570421_kernel<<<grid, block, 0, stream>>>(x, starts, widths, out);
}